// TransformerConv_23390391894661
// MI455X (gfx1250) — compile-verified
//
#include <hip/hip_runtime.h>
#include <hip/hip_bf16.h>
#include <math.h>

// ---------------------------------------------------------------------------
// GAT layer for MI455X (gfx1250, wave32).
//   Stage A (one fused kernel, WMMA f32 16x16x4):
//     fv = feat @ W_val + b_val                    (stored, needed for messages)
//     el = <feat @ W_src + b_src, attn_l> per head (reduced in-register, never stored as fs)
//     er = <feat @ W_dst + b_dst, attn_r> per head
//   Stage B: edge softmax over dst + weighted aggregation (f32 atomics in L2)
// ---------------------------------------------------------------------------

#define GAT_IN   128
#define GAT_H    4
#define GAT_D    32
#define GAT_HD   128
#define GAT_SLOPE 0.2f

typedef float v2f __attribute__((ext_vector_type(2)));
typedef float v8f __attribute__((ext_vector_type(8)));

#define LDS_STRIDE 132   // 128 + 4 pad: row r, col k -> bank (4r+k)%64, conflict-free

// ---------------------------------------------------------------------------
// Init: zero output + logit accumulators, max=-inf, denom=0.
// ---------------------------------------------------------------------------
__global__ __launch_bounds__(256) void gat_init(float* __restrict__ out,
                                                float* __restrict__ el,
                                                float* __restrict__ er,
                                                float* __restrict__ gmax,
                                                float* __restrict__ gden,
                                                int n_nodes) {
    int i = blockIdx.x * blockDim.x + threadIdx.x;
    if (i < n_nodes * GAT_HD) out[i] = 0.0f;
    if (i < n_nodes * GAT_H) {
        el[i] = 0.0f; er[i] = 0.0f;
        gmax[i] = -INFINITY; gden[i] = 0.0f;
    }
}

// ---------------------------------------------------------------------------
// One 16x16 fp32 output tile via 32x V_WMMA_F32_16X16X4_F32.
// Fragment layouts per CDNA5 ISA 7.12.2 (fp32):
//   A 16x4 : lanes 0-15 -> M=lane, v[0]=K0 v[1]=K1 ; lanes 16-31 -> K2,K3
//   B 4x16 : v[0]: lanes0-15 K=0 N=lane / lanes16-31 K=2 ; v[1]: K=1 / K=3
//   C 16x16: VGPR r: lanes0-15 -> M=r, N=lane ; lanes16-31 -> M=r+8
// ---------------------------------------------------------------------------
__device__ __forceinline__ v8f wmma_tile(const float* __restrict__ W,
                                         const float* sA,
                                         int arow, int koff, int col) {
    v8f c = {};
    #pragma unroll
    for (int k = 0; k < GAT_IN; k += 4) {
        v2f a, b;
        a[0] = sA[arow * LDS_STRIDE + k + koff];
        a[1] = sA[arow * LDS_STRIDE + k + koff + 1];
        b[0] = W[(k + koff)     * GAT_HD + col];
        b[1] = W[(k + koff + 1) * GAT_HD + col];
        c = __builtin_amdgcn_wmma_f32_16x16x4_f32(
                /*neg_a=*/false, a, /*neg_b=*/false, b,
                /*c_mod=*/(short)0, c, /*reuse_a=*/false, /*reuse_b=*/false);
    }
    return c;
}

// Reduce a C-fragment tile against an attention vector slice and accumulate
// per-row logits. Within each 16-lane half all lanes share M and enumerate N,
// so a 4-step xor butterfly sums the wave's 16 columns; the two waves that
// cover one head combine through atomicAdd.
__device__ __forceinline__ void logit_reduce(v8f c, float bias, float av,
                                             float* __restrict__ acc,
                                             int orow, int h, int lane) {
    #pragma unroll
    for (int r = 0; r < 8; ++r) {
        float v = (c[r] + bias) * av;
        v += __shfl_xor(v, 1);
        v += __shfl_xor(v, 2);
        v += __shfl_xor(v, 4);
        v += __shfl_xor(v, 8);
        if ((lane & 15) == 0)
            atomicAdd(&acc[(orow + r) * GAT_H + h], v);
    }
}

// ---------------------------------------------------------------------------
// Fused projection kernel. One block = one 16-node row tile in LDS shared by
// 8 waves; wave w owns output columns [16w,16w+16) (= half of head w/2).
// ---------------------------------------------------------------------------
__global__ __launch_bounds__(256) void gat_proj_wmma(
    const float* __restrict__ feat,
    const float* __restrict__ Ws, const float* __restrict__ bs,
    const float* __restrict__ Wd, const float* __restrict__ bd,
    const float* __restrict__ Wv, const float* __restrict__ bv,
    const float* __restrict__ attn_l, const float* __restrict__ attn_r,
    float* __restrict__ el, float* __restrict__ er, float* __restrict__ fv) {

    __shared__ float sA[16 * LDS_STRIDE];

    const int tid  = threadIdx.x;
    const int row0 = blockIdx.x * 16;

    // Cooperative tile load: 16x128 fp32 = 512 float4, 2 per thread.
    {
        const float4* g = (const float4*)(feat + (size_t)row0 * GAT_IN);
        #pragma unroll
        for (int i = tid; i < 16 * 32; i += 256) {
            int r  = i >> 5;
            int c4 = i & 31;
            float4 v = g[r * 32 + c4];
            float* p = &sA[r * LDS_STRIDE + c4 * 4];
            p[0] = v.x; p[1] = v.y; p[2] = v.z; p[3] = v.w;
        }
    }
    __syncthreads();

    const int wave = tid >> 5;            // 0..7 -> column tile
    const int lane = tid & 31;
    const int ncol = lane & 15;
    const int col  = wave * 16 + ncol;    // output column 0..127
    const int koff = (lane >> 4) * 2;     // 0 for lanes 0-15, 2 for lanes 16-31
    const int arow = lane & 15;           // A fragment row
    const int h    = wave >> 1;           // head this column tile belongs to
    const int dcol = (wave & 1) * 16 + ncol;   // d index within head
    const int orow = row0 + (lane >> 4) * 8;   // base output row for C frag

    // ---- fs tile -> el (never stored) ----
    {
        v8f c = wmma_tile(Ws, sA, arow, koff, col);
        logit_reduce(c, bs[col], attn_l[h * GAT_D + dcol], el, orow, h, lane);
    }
    // ---- fd tile -> er (never stored) ----
    {
        v8f c = wmma_tile(Wd, sA, arow, koff, col);
        logit_reduce(c, bd[col], attn_r[h * GAT_D + dcol], er, orow, h, lane);
    }
    // ---- fv tile -> stored for message aggregation ----
    {
        v8f c = wmma_tile(Wv, sA, arow, koff, col);
        const float bias = bv[col];
        #pragma unroll
        for (int r = 0; r < 8; ++r)
            fv[(size_t)(orow + r) * GAT_HD + col] = c[r] + bias;
    }
}

// Float atomic max via sign-split int trick (valid with -inf init).
__device__ __forceinline__ void atomicMaxF(float* addr, float v) {
    if (v >= 0.0f) atomicMax((int*)addr, __float_as_int(v));
    else           atomicMin((unsigned int*)addr, __float_as_uint(v));
}

__device__ __forceinline__ float leaky(float e) {
    return e >= 0.0f ? e : GAT_SLOPE * e;
}

// ---------------------------------------------------------------------------
// Edge pass 1: running max of leaky(el[src]+er[dst]) per (dst,head).
// ---------------------------------------------------------------------------
__global__ __launch_bounds__(256) void gat_edge_max(
    const float* __restrict__ el, const float* __restrict__ er,
    const int* __restrict__ src, const int* __restrict__ dst,
    float* __restrict__ gmax, int n_edges) {
    int idx = blockIdx.x * blockDim.x + threadIdx.x;   // (edge, h)
    if (idx >= n_edges * GAT_H) return;
    int h = idx & (GAT_H - 1);
    int e = idx >> 2;
    float v = leaky(el[src[e] * GAT_H + h] + er[dst[e] * GAT_H + h]);
    atomicMaxF(&gmax[dst[e] * GAT_H + h], v);
}

// ---------------------------------------------------------------------------
// Edge pass 2: ee = exp(e - max[dst]); accumulate denom.
// ---------------------------------------------------------------------------
__global__ __launch_bounds__(256) void gat_edge_exp(
    const float* __restrict__ el, const float* __restrict__ er,
    const int* __restrict__ src, const int* __restrict__ dst,
    const float* __restrict__ gmax, float* __restrict__ gden,
    float* __restrict__ ee, int n_edges) {
    int idx = blockIdx.x * blockDim.x + threadIdx.x;
    if (idx >= n_edges * GAT_H) return;
    int h = idx & (GAT_H - 1);
    int e = idx >> 2;
    int dh = dst[e] * GAT_H + h;
    float v  = leaky(el[src[e] * GAT_H + h] + er[dh]);
    float ex = __expf(v - gmax[dh]);
    ee[idx] = ex;
    atomicAdd(&gden[dh], ex);
}

// ---------------------------------------------------------------------------
// Edge pass 3: normalize attention coefficients in place.
// ---------------------------------------------------------------------------
__global__ __launch_bounds__(256) void gat_edge_norm(
    const int* __restrict__ dst, const float* __restrict__ gden,
    float* __restrict__ ee, int n_edges) {
    int idx = blockIdx.x * blockDim.x + threadIdx.x;
    if (idx >= n_edges * GAT_H) return;
    int h = idx & (GAT_H - 1);
    int e = idx >> 2;
    ee[idx] = ee[idx] / gden[dst[e] * GAT_H + h];
}

// ---------------------------------------------------------------------------
// Edge pass 4: out[dst, h*32+d] += a[edge,h] * fv[src, h*32+d].
// One thread per (edge, element): 32 consecutive lanes share one coefficient,
// coalesced 128B gather from fv and 128B of f32 atomics into L2-resident out.
// ---------------------------------------------------------------------------
__global__ __launch_bounds__(256) void gat_aggregate(
    const float* __restrict__ fv, const float* __restrict__ a,
    const int* __restrict__ src, const int* __restrict__ dst,
    float* __restrict__ out, int n_edges) {
    unsigned int gid = blockIdx.x * 256u + threadIdx.x;
    if (gid >= (unsigned int)n_edges * GAT_HD) return;
    unsigned int e = gid >> 7;          // edge
    unsigned int r = gid & 127;         // element within H*D row
    unsigned int h = r >> 5;            // head
    float coef = a[e * GAT_H + h];
    float val  = fv[(size_t)src[e] * GAT_HD + r];
    atomicAdd(&out[(size_t)dst[e] * GAT_HD + r], coef * val);
}

// ---------------------------------------------------------------------------
// Launch
// ---------------------------------------------------------------------------
extern "C" void kernel_launch(void* const* d_in, const int* in_sizes, int n_in,
                              void* d_out, int out_size, void* d_ws, size_t ws_size,
                              hipStream_t stream) {
    const float* feat   = (const float*)d_in[0];
    const float* W_src  = (const float*)d_in[1];
    const float* b_src  = (const float*)d_in[2];
    const float* W_dst  = (const float*)d_in[3];
    const float* b_dst  = (const float*)d_in[4];
    const float* W_val  = (const float*)d_in[5];
    const float* b_val  = (const float*)d_in[6];
    const float* attn_l = (const float*)d_in[7];
    const float* attn_r = (const float*)d_in[8];
    const int*   src    = (const int*)d_in[9];
    const int*   dst    = (const int*)d_in[10];

    const int N = in_sizes[0] / GAT_IN;   // 50000 (= 3125 * 16)
    const int E = in_sizes[9];            // 800000

    float* out = (float*)d_out;

    // Workspace carve-up (fp32): fv, el, er, max, denom, ee  (~42 MB)
    float* ws   = (float*)d_ws;
    float* fv   = ws;
    float* el   = fv   + (size_t)N * GAT_HD;
    float* er   = el   + (size_t)N * GAT_H;
    float* gmax = er   + (size_t)N * GAT_H;
    float* gden = gmax + (size_t)N * GAT_H;
    float* ee   = gden + (size_t)N * GAT_H;

    // 1) init accumulators
    {
        int total = N * GAT_HD;
        gat_init<<<(total + 255) / 256, 256, 0, stream>>>(out, el, er, gmax, gden, N);
    }
    // 2) fused projections + logits
    {
        gat_proj_wmma<<<N / 16, 256, 0, stream>>>(
            feat, W_src, b_src, W_dst, b_dst, W_val, b_val,
            attn_l, attn_r, el, er, fv);
    }
    // 3) edge softmax
    {
        int total = E * GAT_H;
        int blocks = (total + 255) / 256;
        gat_edge_max <<<blocks, 256, 0, stream>>>(el, er, src, dst, gmax, E);
        gat_edge_exp <<<blocks, 256, 0, stream>>>(el, er, src, dst, gmax, gden, ee, E);
        gat_edge_norm<<<blocks, 256, 0, stream>>>(dst, gden, ee, E);
    }
    // 4) aggregate
    {
        unsigned int total = (unsigned int)E * GAT_HD;
        gat_aggregate<<<(total + 255u) / 256u, 256, 0, stream>>>(
            fv, ee, src, dst, out, E);
    }
}